// Phoneme_SSL_loss_4294967296199
// MI455X (gfx1250) — compile-verified
//
#include <hip/hip_runtime.h>
#include <hip/hip_bf16.h>
#include <math.h>

// Problem constants from the reference
#define B_   32
#define T_   128
#define NF_  20
#define D_   512
#define S_   5
#define L_   96
#define NPOS (B_ * L_)    // 3072 gathered positions
#define NA   (NF_ - 1)    // 19 anchors
#define EPS_ 1e-8f

typedef __attribute__((ext_vector_type(2))) float v2f;
typedef __attribute__((ext_vector_type(4))) float v4f;
typedef __attribute__((ext_vector_type(8))) float v8f;

// Padded LDS row stride (floats): 512 + 4 -> strided fragment reads are bank-conflict-free
#define LDS_STRIDE 516
#define NROWS      21                        // 20 data rows + 1 zero pad row (row 20)
#define TILE_FLOATS (NROWS * LDS_STRIDE)     // 10836 floats = 43344 bytes

__device__ __forceinline__ unsigned lds_byte_offset(const void* p) {
  // generic -> LDS(AS3) addrspace cast; AS3 pointers are 32-bit LDS byte offsets
  return (unsigned)(size_t)(__attribute__((address_space(3))) const char*)p;
}

__global__ __launch_bounds__(256)
void gram_loss_kernel(const float* __restrict__ out4d,
                      const int*   __restrict__ batch_idx,
                      const int*   __restrict__ time_idx,
                      const int*   __restrict__ neg_idx,
                      float*       __restrict__ pos_loss) {
  __shared__ float smem[TILE_FLOATS];

  const int n   = blockIdx.x;       // position index (0..3071)
  const int tid = threadIdx.x;
  const int b   = batch_idx[n];
  const int t   = time_idx[n];
  const float* src = out4d + ((size_t)(b * T_ + t)) * (size_t)(NF_ * D_);

  // ---- Zero the pad row (row 20) so the upper Gram tile can load branch-free ----
  for (int i = tid; i < LDS_STRIDE; i += 256) smem[20 * LDS_STRIDE + i] = 0.f;

  // ---- Stage 20x512 f32 tile into LDS via async global->LDS b128 (ASYNCcnt path) ----
  {
    const unsigned base = lds_byte_offset(smem);
    for (int idx = tid; idx < NF_ * (D_ / 4); idx += 256) {
      const int row = idx >> 7;     // 128 float4 per row
      const int c4  = idx & 127;
      const float* g = src + row * D_ + c4 * 4;
      const unsigned lo = base + (unsigned)((row * LDS_STRIDE + c4 * 4) * 4);
      asm volatile("global_load_async_to_lds_b128 %0, %1, off"
                   :: "v"(lo), "v"(g) : "memory");
    }
    asm volatile("s_wait_asynccnt 0x0" ::: "memory");
  }
  __syncthreads();

  // ---- Gram matrix G = F * F^T via V_WMMA_F32_16X16X4_F32 ----
  // A-frag layout (16x4 f32): lane L (L<16): v0=F[L][k0], v1=F[L][k0+1];
  //                           lane L+16   : v0=F[L][k0+2], v1=F[L][k0+3].
  // Passing the same registers as B yields B = F^T tile (standard X*X^T idiom).
  const int wave = tid >> 5;
  const int lane = tid & 31;
  const int half = lane >> 4;
  const int r    = lane & 15;
  const int r2   = (r < (NF_ - 16)) ? (16 + r) : 20;   // row 20 = zeros

  v8f acc00 = {}; v8f acc01 = {}; v8f acc10 = {}; v8f acc11 = {};

  const int kbase = wave * 64;      // 8 waves split K=512
  #pragma unroll
  for (int kk = 0; kk < 16; ++kk) {
    const int col = kbase + kk * 4 + 2 * half;
    v2f a0 = *(const v2f*)(&smem[r  * LDS_STRIDE + col]);   // rows 0..15
    v2f a1 = *(const v2f*)(&smem[r2 * LDS_STRIDE + col]);   // rows 16..19 (+zeros)
    acc00 = __builtin_amdgcn_wmma_f32_16x16x4_f32(false, a0, false, a0, (short)0, acc00, false, false);
    acc01 = __builtin_amdgcn_wmma_f32_16x16x4_f32(false, a0, false, a1, (short)0, acc01, false, false);
    acc10 = __builtin_amdgcn_wmma_f32_16x16x4_f32(false, a1, false, a0, (short)0, acc10, false, false);
    acc11 = __builtin_amdgcn_wmma_f32_16x16x4_f32(false, a1, false, a1, (short)0, acc11, false, false);
  }
  __syncthreads();   // all waves done reading the tile; reuse smem below

  // ---- Cross-wave reduction of partial Grams (accbuf = smem[0..8192)) ----
  {
    float* dst = smem + (wave * 32 + lane) * 32;
    #pragma unroll
    for (int v = 0; v < 8; ++v) {
      dst[0 * 8 + v] = acc00[v];
      dst[1 * 8 + v] = acc01[v];
      dst[2 * 8 + v] = acc10[v];
      dst[3 * 8 + v] = acc11[v];
    }
  }
  __syncthreads();

  float* Gm = smem + 8192;          // 20 x 21 floats, padded
  for (int e = tid; e < 1024; e += 256) {
    const int Ln   = e >> 5;
    const int slot = e & 31;
    float sum = 0.f;
    #pragma unroll
    for (int w = 0; w < 8; ++w) sum += smem[w * 1024 + e];
    // C/D layout: VGPR v, lanes 0-15 -> M=v, lanes 16-31 -> M=v+8; N = lane&15
    const int tile = slot >> 3;
    const int v    = slot & 7;
    const int row  = (tile >> 1) * 16 + v + 8 * (Ln >> 4);
    const int colg = (tile & 1) * 16 + (Ln & 15);
    if (row < NF_ && colg < NF_) Gm[row * (NF_ + 1) + colg] = sum;
  }
  __syncthreads();

  // ---- Cosine sims + log-softmax per anchor ----
  float* lossbuf = smem + 8192 + NF_ * (NF_ + 1);
  if (tid < NA) {
    const int a = tid;
    const float sa = sqrtf(Gm[a * (NF_ + 1) + a]);
    float s[1 + S_];
    {
      const int j = a + 1;
      const float den = fmaxf(sa * sqrtf(Gm[j * (NF_ + 1) + j]), EPS_);
      s[0] = Gm[a * (NF_ + 1) + j] / den;
    }
    #pragma unroll
    for (int i = 0; i < S_; ++i) {
      const int j = neg_idx[a * S_ + i];
      const float den = fmaxf(sa * sqrtf(Gm[j * (NF_ + 1) + j]), EPS_);
      s[1 + i] = Gm[a * (NF_ + 1) + j] / den;
    }
    float m = s[0];
    #pragma unroll
    for (int i = 1; i <= S_; ++i) m = fmaxf(m, s[i]);
    float es = 0.f;
    #pragma unroll
    for (int i = 0; i <= S_; ++i) es += expf(s[i] - m);
    lossbuf[a] = (m + logf(es)) - s[0];   // -log_softmax[...,0]
  }
  __syncthreads();

  if (tid == 0) {
    float tot = 0.f;
    #pragma unroll
    for (int a = 0; a < NA; ++a) tot += lossbuf[a];
    pos_loss[n] = tot;
  }
}

// Deterministic final reduction: sum 3072 partial losses, divide by (3072*19)
__global__ __launch_bounds__(256)
void reduce_loss_kernel(const float* __restrict__ pos_loss, float* __restrict__ out) {
  __shared__ float red[256];
  const int tid = threadIdx.x;
  float s = 0.f;
  for (int i = tid; i < NPOS; i += 256) s += pos_loss[i];
  red[tid] = s;
  __syncthreads();
  for (int off = 128; off > 0; off >>= 1) {
    if (tid < off) red[tid] += red[tid + off];
    __syncthreads();
  }
  if (tid == 0) out[0] = red[0] / (float)(NPOS * NA);
}

extern "C" void kernel_launch(void* const* d_in, const int* in_sizes, int n_in,
                              void* d_out, int out_size, void* d_ws, size_t ws_size,
                              hipStream_t stream) {
  (void)in_sizes; (void)n_in; (void)out_size; (void)ws_size;
  const float* out4d     = (const float*)d_in[0];
  const int*   batch_idx = (const int*)d_in[1];
  const int*   time_idx  = (const int*)d_in[2];
  const int*   neg_idx   = (const int*)d_in[3];
  float*       ws        = (float*)d_ws;          // NPOS floats of scratch

  gram_loss_kernel<<<dim3(NPOS), dim3(256), 0, stream>>>(out4d, batch_idx, time_idx, neg_idx, ws);
  reduce_loss_kernel<<<dim3(1), dim3(256), 0, stream>>>(ws, (float*)d_out);
}